// MultiModalGraphAttention_56435870270026
// MI455X (gfx1250) — compile-verified
//
#include <hip/hip_runtime.h>
#include <hip/hip_bf16.h>

// ---------------------------------------------------------------------------
// MultiModalGraphAttention forward for MI455X (gfx1250, wave32).
// GEMMs use V_WMMA_F32_16X16X4_F32 (native fp32 matrix op -> exact fp32 math)
// with async global->LDS staging of the B tile (ASYNCcnt / s_wait_asynccnt).
// Graph attention (segment softmax + scatter) uses L2 float atomics.
// ---------------------------------------------------------------------------

typedef float v2f __attribute__((ext_vector_type(2)));
typedef float v8f __attribute__((ext_vector_type(8)));

#define N_PER   10000
#define N_MOD   3
#define NNODES  30000
#define NEDGES  240000
#define HEADS   8
#define HID     128
#define NEG_SLOPE 0.2f

// B-tile LDS row stride (floats): 64 data + 8 pad so that the two half-wave
// row offsets (rows k and k+2) land on disjoint bank groups (2*72 % 64 == 16).
#define BROW 72

static inline int ceildiv(int a, int b) { return (a + b - 1) / b; }

// -------- CDNA5 async global->LDS copy (16B per lane) + counter wait --------
__device__ inline void async_copy_b128(unsigned lds_off, const float* gptr) {
    asm volatile("global_load_async_to_lds_b128 %0, %1, off"
                 :: "v"(lds_off), "v"(gptr)
                 : "memory");
}
__device__ inline void wait_async0() {
    asm volatile("s_wait_asynccnt 0x0" ::: "memory");
}

// Stage a 16(K) x 64(N) chunk of W into LDS buf[16][BROW] (4KB of data).
// 128 threads x 2 iterations x 16B = 4KB.  Wchunk points at W[kbase][colBase].
__device__ inline void stage_b_chunk(const float* __restrict__ Wchunk, int N,
                                     float (*buf)[BROW], int t) {
#pragma unroll
    for (int r = 0; r < 2; ++r) {
        int idx = t + r * 128;          // 0..255 -> (row, 16B chunk-in-row)
        int row = idx >> 4;             // 16 rows
        int c16 = idx & 15;             // 16 chunks of 4 floats per row
        const float* g = Wchunk + (size_t)row * N + c16 * 4;
        unsigned lds = (unsigned)(size_t)&buf[row][c16 * 4];
        async_copy_b128(lds, g);
    }
}

// ----------------------------------------------------------------------------
// WMMA fp32 GEMM:  C[M,N] = act(A[M,K] @ W[K,N] (+ bias[N]))
// block = 128 threads = 4 waves; each wave computes a 32(M) x 64(N) tile with
// 8 accumulators of v_wmma_f32_16x16x4_f32; block tile = 128 x 64.
// B chunks (16 x 64) are async-staged to LDS, double buffered; all 4 waves
// share them.  A fragments are contiguous b64 global loads (per-wave rows).
// Rows are clamped on load / guarded on store so EXEC stays all-ones for WMMA.
// Requires K % 16 == 0, N % 64 == 0 (all shapes here satisfy this).
// ----------------------------------------------------------------------------
template <bool RELU, bool BIAS>
__global__ __launch_bounds__(128)
void gemm_wmma_f32(const float* __restrict__ A, const float* __restrict__ W,
                   const float* __restrict__ bias, float* __restrict__ C,
                   int M, int K, int N)
{
    __shared__ float bs[2][16][BROW];

    const int lane    = threadIdx.x & 31;
    const int wave    = threadIdx.x >> 5;
    const int rowBase = blockIdx.y * 128 + wave * 32;  // 2 x 16-row subtiles
    const int colBase = blockIdx.x * 64;               // 4 x 16-col subtiles

    // fp32 16x4 A-fragment: lanes 0-15 -> K = k+0,k+1 ; lanes 16-31 -> k+2,k+3
    const int am = lane & 15;
    const int ak = (lane >> 4) << 1;   // 0 or 2
    // fp32 4x16 B-fragment: lanes 0-15 -> K rows k+0/k+1 ; lanes 16-31 -> k+2/k+3
    const int bn = lane & 15;
    const int bk = (lane >> 4) << 1;   // 0 or 2

    int r0 = rowBase + am;
    int r1 = rowBase + 16 + am;
    const int r0c = (r0 < M) ? r0 : (M - 1);
    const int r1c = (r1 < M) ? r1 : (M - 1);

    v8f acc[2][4];
#pragma unroll
    for (int i = 0; i < 2; ++i)
#pragma unroll
        for (int j = 0; j < 4; ++j)
            acc[i][j] = {};

    const float* a0base = A + (size_t)r0c * K + ak;
    const float* a1base = A + (size_t)r1c * K + ak;

    const int nchunks = K >> 4;                 // K/16
    const float* wchunk = W + colBase;          // W[kbase=0][colBase]
    const size_t wstep  = (size_t)16 * N;

    // prologue: stage chunk 0 into buffer 0
    stage_b_chunk(wchunk, N, bs[0], threadIdx.x);
    wait_async0();
    __syncthreads();

    for (int c = 0; c < nchunks; ++c) {
        const int cur = c & 1;
        if (c + 1 < nchunks)
            stage_b_chunk(wchunk + wstep * (c + 1), N, bs[cur ^ 1], threadIdx.x);

        const int kbase = c << 4;
#pragma unroll
        for (int kk = 0; kk < 16; kk += 4) {
            v2f a0, a1;
            const float* ap0 = a0base + kbase + kk;
            a0.x = ap0[0];  a0.y = ap0[1];
            const float* ap1 = a1base + kbase + kk;
            a1.x = ap1[0];  a1.y = ap1[1];
#pragma unroll
            for (int j = 0; j < 4; ++j) {
                v2f b;
                b.x = bs[cur][kk + bk][j * 16 + bn];
                b.y = bs[cur][kk + bk + 1][j * 16 + bn];
                acc[0][j] = __builtin_amdgcn_wmma_f32_16x16x4_f32(
                    false, a0, false, b, (short)0, acc[0][j], false, false);
                acc[1][j] = __builtin_amdgcn_wmma_f32_16x16x4_f32(
                    false, a1, false, b, (short)0, acc[1][j], false, false);
            }
        }
        // publish staged chunk c+1 and make sure all waves finished reading
        // buffer `cur` before iteration c+1 overwrites it.
        wait_async0();
        __syncthreads();
    }

    // C/D layout: VGPR e -> M = e (lanes 0-15) / 8+e (lanes 16-31); N = lane&15
    const int mof = (lane >> 4) * 8;
    const int cn0 = colBase + (lane & 15);
#pragma unroll
    for (int i = 0; i < 2; ++i) {
#pragma unroll
        for (int j = 0; j < 4; ++j) {
            const int n = cn0 + j * 16;
#pragma unroll
            for (int e = 0; e < 8; ++e) {
                int m = rowBase + i * 16 + mof + e;
                if (m < M) {
                    float v = acc[i][j][e];
                    if (BIAS) v += bias[n];
                    if (RELU) v = fmaxf(v, 0.0f);
                    C[(size_t)m * N + n] = v;
                }
            }
        }
    }
}

// ---------------- order-preserving float<->uint encode (for atomicMax) ------
__device__ inline unsigned fenc(float f) {
    unsigned u = __float_as_uint(f);
    return (u & 0x80000000u) ? ~u : (u | 0x80000000u);
}
__device__ inline float fdec(unsigned u) {
    return (u & 0x80000000u) ? __uint_as_float(u & 0x7FFFFFFFu)
                             : __uint_as_float(~u);
}
#define ENC_NEG_INF 0x007FFFFFu  // fenc(-inf)

__global__ void init_mz_kernel(unsigned* __restrict__ menc,
                               float* __restrict__ z, int n)
{
    int t = blockIdx.x * blockDim.x + threadIdx.x;
    if (t < n) { menc[t] = ENC_NEG_INF; z[t] = 0.0f; }
}

// alpha_src[n,h] = sum_c h[n,h,c]*a_src[h,c]  (and same for dst); wave/(n,h)
__global__ __launch_bounds__(256)
void alpha_kernel(const float* __restrict__ hfeat,
                  const float* __restrict__ a_src, const float* __restrict__ a_dst,
                  float* __restrict__ as, float* __restrict__ ad,
                  int Nn, int H, int C)
{
    int wid  = (int)((blockIdx.x * blockDim.x + threadIdx.x) >> 5);
    int lane = threadIdx.x & 31;
    if (wid >= Nn * H) return;
    int n = wid / H, h = wid % H;
    const float* hp = hfeat + (size_t)n * (H * C) + (size_t)h * C;
    float ss = 0.0f, sd = 0.0f;
    for (int c = lane; c < C; c += 32) {
        float x = hp[c];
        ss += x * a_src[h * C + c];
        sd += x * a_dst[h * C + c];
    }
#pragma unroll
    for (int o = 16; o; o >>= 1) {
        ss += __shfl_xor(ss, o, 32);
        sd += __shfl_xor(sd, o, 32);
    }
    if (lane == 0) { as[n * H + h] = ss; ad[n * H + h] = sd; }
}

// pass 1: segment max of leaky_relu(e) over dst
__global__ void edge_max_kernel(const int* __restrict__ ei,
                                const float* __restrict__ as, const float* __restrict__ ad,
                                unsigned* __restrict__ menc, int E, int H)
{
    int t = blockIdx.x * blockDim.x + threadIdx.x;
    if (t >= E * H) return;
    int e = t / H, h = t % H;
    int s = ei[e], d = ei[E + e];
    float v = as[s * H + h] + ad[d * H + h];
    v = (v > 0.0f) ? v : NEG_SLOPE * v;
    atomicMax(menc + d * H + h, fenc(v));
}

// pass 2: segment sum of exp(e - m)
__global__ void edge_sum_kernel(const int* __restrict__ ei,
                                const float* __restrict__ as, const float* __restrict__ ad,
                                const unsigned* __restrict__ menc,
                                float* __restrict__ z, int E, int H)
{
    int t = blockIdx.x * blockDim.x + threadIdx.x;
    if (t >= E * H) return;
    int e = t / H, h = t % H;
    int s = ei[e], d = ei[E + e];
    float v = as[s * H + h] + ad[d * H + h];
    v = (v > 0.0f) ? v : NEG_SLOPE * v;
    float m = fdec(menc[d * H + h]);
    atomicAdd(z + d * H + h, __expf(v - m));
}

// pass 3: agg[d,h,:] += a * h[s,h,:]   (wave per (edge,head), 4 ch per lane)
__global__ __launch_bounds__(256)
void edge_agg_kernel(const int* __restrict__ ei,
                     const float* __restrict__ as, const float* __restrict__ ad,
                     const unsigned* __restrict__ menc, const float* __restrict__ z,
                     const float* __restrict__ hfeat, float* __restrict__ agg,
                     int E, int H, int C)
{
    int wid  = (int)((blockIdx.x * blockDim.x + threadIdx.x) >> 5);
    int lane = threadIdx.x & 31;
    if (wid >= E * H) return;
    int e = wid / H, h = wid % H;
    int s = ei[e], d = ei[E + e];
    float v = as[s * H + h] + ad[d * H + h];
    v = (v > 0.0f) ? v : NEG_SLOPE * v;
    float m = fdec(menc[d * H + h]);
    float a = __expf(v - m) / (z[d * H + h] + 1e-16f);
    const float* hs = hfeat + (size_t)s * (H * C) + (size_t)h * C;
    float* ag = agg + (size_t)d * (H * C) + (size_t)h * C;
    for (int c = lane; c < C; c += 32)
        atomicAdd(ag + c, hs[c] * a);
}

// in-place: x = relu(x + b[col])
__global__ void finalize_kernel(float* __restrict__ x, const float* __restrict__ b,
                                long long total, int F)
{
    long long t = (long long)blockIdx.x * blockDim.x + threadIdx.x;
    if (t >= total) return;
    int col = (int)(t % F);
    float v = x[t] + b[col];
    x[t] = (v > 0.0f) ? v : 0.0f;
}

// mean-pool per modality: pooled[mod][c] = mean_n x[mod*per+n][c]
__global__ __launch_bounds__(128)
void pool_kernel(const float* __restrict__ x, float* __restrict__ pooled, int per)
{
    int mod = blockIdx.x, c = threadIdx.x;
    const float* xp = x + (size_t)mod * per * HID;
    float s = 0.0f;
    for (int n = 0; n < per; ++n) s += xp[(size_t)n * HID + c];
    pooled[mod * HID + c] = s / (float)per;
}

// cross-attention (3 tokens, 8 heads, hd=16) + fusion MLP -> out[128]
__global__ __launch_bounds__(128)
void head_kernel(const float* __restrict__ pooled,
                 const float* __restrict__ Win, const float* __restrict__ bin,
                 const float* __restrict__ Wout, const float* __restrict__ bout,
                 const float* __restrict__ W1, const float* __restrict__ b1,
                 const float* __restrict__ W2, const float* __restrict__ b2,
                 float* __restrict__ out)
{
    __shared__ float p[3][HID], q[3][HID], kk[3][HID], v[3][HID];
    __shared__ float fused[3 * HID], hh[HID];
    int j = threadIdx.x;
    for (int i = 0; i < 3; ++i) p[i][j] = pooled[i * HID + j];
    __syncthreads();
    // qkv = pooled @ Win.T + bin ; Win is [384,128]
    for (int i = 0; i < 3; ++i) {
        float sq = bin[j], sk = bin[HID + j], sv = bin[2 * HID + j];
        for (int c = 0; c < HID; ++c) {
            float pc = p[i][c];
            sq += pc * Win[(size_t)j * HID + c];
            sk += pc * Win[(size_t)(HID + j) * HID + c];
            sv += pc * Win[(size_t)(2 * HID + j) * HID + c];
        }
        q[i][j] = sq; kk[i][j] = sk; v[i][j] = sv;
    }
    __syncthreads();
    __shared__ float o[3][HID];
    const int hq = j >> 4;  // head (hd = 16)
    for (int i = 0; i < 3; ++i) {
        float l[3];
        for (int t = 0; t < 3; ++t) {
            float s = 0.0f;
            for (int dd = 0; dd < 16; ++dd) s += q[i][hq * 16 + dd] * kk[t][hq * 16 + dd];
            l[t] = s * 0.25f;  // 1/sqrt(16)
        }
        float mx = fmaxf(l[0], fmaxf(l[1], l[2]));
        float e0 = __expf(l[0] - mx), e1 = __expf(l[1] - mx), e2 = __expf(l[2] - mx);
        float inv = 1.0f / (e0 + e1 + e2);
        o[i][j] = (e0 * v[0][j] + e1 * v[1][j] + e2 * v[2][j]) * inv;
    }
    __syncthreads();
    // out-proj: fused[i*128+j] = sum_c o[i][c]*Wout[j][c] + bout[j]
    for (int i = 0; i < 3; ++i) {
        float s = bout[j];
        for (int c = 0; c < HID; ++c) s += o[i][c] * Wout[(size_t)j * HID + c];
        fused[i * HID + j] = s;
    }
    __syncthreads();
    // fusion l1: relu(fused @ W1 + b1), W1 is [384,128] (x @ W convention)
    float s = b1[j];
    for (int c = 0; c < 3 * HID; ++c) s += fused[c] * W1[(size_t)c * HID + j];
    hh[j] = (s > 0.0f) ? s : 0.0f;
    __syncthreads();
    float s2 = b2[j];
    for (int c = 0; c < HID; ++c) s2 += hh[c] * W2[(size_t)c * HID + j];
    out[j] = s2;
}

// ---------------------------------------------------------------------------
// Host-side orchestration
// ---------------------------------------------------------------------------
static void run_gemm(const float* A, const float* W, const float* bias, float* C,
                     int M, int K, int N, bool relu, bool hasBias, hipStream_t s)
{
    dim3 grid(N / 64, ceildiv(M, 128));
    if (hasBias) {
        if (relu) gemm_wmma_f32<true,  true ><<<grid, 128, 0, s>>>(A, W, bias, C, M, K, N);
        else      gemm_wmma_f32<false, true ><<<grid, 128, 0, s>>>(A, W, bias, C, M, K, N);
    } else {
        if (relu) gemm_wmma_f32<true,  false><<<grid, 128, 0, s>>>(A, W, bias, C, M, K, N);
        else      gemm_wmma_f32<false, false><<<grid, 128, 0, s>>>(A, W, bias, C, M, K, N);
    }
}

static void run_gat_layer(const int* ei, const float* hfeat, float* agg,
                          const float* a_src, const float* a_dst, const float* b,
                          float* as, float* ad, unsigned* menc, float* z,
                          int H, int C, hipStream_t s)
{
    const int NH = NNODES * H;
    init_mz_kernel<<<ceildiv(NH, 256), 256, 0, s>>>(menc, z, NH);
    alpha_kernel<<<ceildiv(NH, 8), 256, 0, s>>>(hfeat, a_src, a_dst, as, ad, NNODES, H, C);
    const int EH = NEDGES * H;
    edge_max_kernel<<<ceildiv(EH, 256), 256, 0, s>>>(ei, as, ad, menc, NEDGES, H);
    edge_sum_kernel<<<ceildiv(EH, 256), 256, 0, s>>>(ei, as, ad, menc, z, NEDGES, H);
    hipMemsetAsync(agg, 0, (size_t)NNODES * H * C * sizeof(float), s);
    edge_agg_kernel<<<ceildiv(EH, 8), 256, 0, s>>>(ei, as, ad, menc, z, hfeat, agg, NEDGES, H, C);
    long long total = (long long)NNODES * H * C;
    finalize_kernel<<<(int)((total + 255) / 256), 256, 0, s>>>(agg, b, total, H * C);
}

extern "C" void kernel_launch(void* const* d_in, const int* in_sizes, int n_in,
                              void* d_out, int out_size, void* d_ws, size_t ws_size,
                              hipStream_t stream)
{
    // --- inputs (jax pytree flattening: dicts by sorted keys) ---
    const float* x_power = (const float*)d_in[0];   // [10000,512]
    const float* x_em    = (const float*)d_in[1];   // [10000,256]
    const float* x_tim   = (const float*)d_in[2];   // [10000,128]
    const int*   ei      = (const int*)  d_in[3];   // [2,240000]
    // params: attn{Win,Wout,bin,bout}, enc{em,power,timing}{l1,l2}{W,b},
    //         fus{l1,l2}{W,b}, gat[0..2]{W,a_dst,a_src,b}
    const float* attn_Win  = (const float*)d_in[4];
    const float* attn_Wout = (const float*)d_in[5];
    const float* attn_bin  = (const float*)d_in[6];
    const float* attn_bout = (const float*)d_in[7];
    const float* em_W1  = (const float*)d_in[8];
    const float* em_b1  = (const float*)d_in[9];
    const float* em_W2  = (const float*)d_in[10];
    const float* em_b2  = (const float*)d_in[11];
    const float* pw_W1  = (const float*)d_in[12];
    const float* pw_b1  = (const float*)d_in[13];
    const float* pw_W2  = (const float*)d_in[14];
    const float* pw_b2  = (const float*)d_in[15];
    const float* tm_W1  = (const float*)d_in[16];
    const float* tm_b1  = (const float*)d_in[17];
    const float* tm_W2  = (const float*)d_in[18];
    const float* tm_b2  = (const float*)d_in[19];
    const float* fus_W1 = (const float*)d_in[20];
    const float* fus_b1 = (const float*)d_in[21];
    const float* fus_W2 = (const float*)d_in[22];
    const float* fus_b2 = (const float*)d_in[23];
    const float* g0_W = (const float*)d_in[24];  const float* g0_ad = (const float*)d_in[25];
    const float* g0_as = (const float*)d_in[26]; const float* g0_b  = (const float*)d_in[27];
    const float* g1_W = (const float*)d_in[28];  const float* g1_ad = (const float*)d_in[29];
    const float* g1_as = (const float*)d_in[30]; const float* g1_b  = (const float*)d_in[31];
    const float* g2_W = (const float*)d_in[32];  const float* g2_ad = (const float*)d_in[33];
    const float* g2_as = (const float*)d_in[34]; const float* g2_b  = (const float*)d_in[35];

    // --- workspace layout (~250 MB) ---
    float* bufP = (float*)d_ws;                          // [30000,1024] node feats / agg
    float* bufQ = bufP + (size_t)NNODES * 1024;          // [30000,1024] GEMM out / enc hidden
    float* as   = bufQ + (size_t)NNODES * 1024;          // [30000,8]
    float* ad   = as + (size_t)NNODES * HEADS;           // [30000,8]
    unsigned* menc = (unsigned*)(ad + (size_t)NNODES * HEADS);  // [30000,8]
    float* zbuf = (float*)(menc + (size_t)NNODES * HEADS);      // [30000,8]
    float* pooled = zbuf + (size_t)NNODES * HEADS;              // [3,128]
    float* outF = (float*)d_out;                                // [128]

    // --- encoders: x = concat(power, em, timing) -> bufP[30000,128] ---
    run_gemm(x_power, pw_W1, pw_b1, bufQ, N_PER, 512, HID, true,  true, stream);
    run_gemm(bufQ,    pw_W2, pw_b2, bufP + 0,                   N_PER, HID, HID, false, true, stream);
    run_gemm(x_em,    em_W1, em_b1, bufQ, N_PER, 256, HID, true,  true, stream);
    run_gemm(bufQ,    em_W2, em_b2, bufP + (size_t)N_PER*HID,   N_PER, HID, HID, false, true, stream);
    run_gemm(x_tim,   tm_W1, tm_b1, bufQ, N_PER, 128, HID, true,  true, stream);
    run_gemm(bufQ,    tm_W2, tm_b2, bufP + (size_t)2*N_PER*HID, N_PER, HID, HID, false, true, stream);

    // --- GAT layer 0: [30000,128] -> h[30000,1024] -> agg/relu -> bufP ---
    run_gemm(bufP, g0_W, nullptr, bufQ, NNODES, 128, 1024, false, false, stream);
    run_gat_layer(ei, bufQ, bufP, g0_as, g0_ad, g0_b, as, ad, menc, zbuf, HEADS, HID, stream);

    // --- GAT layer 1: [30000,1024] -> h[30000,1024] -> agg/relu -> bufP ---
    run_gemm(bufP, g1_W, nullptr, bufQ, NNODES, 1024, 1024, false, false, stream);
    run_gat_layer(ei, bufQ, bufP, g1_as, g1_ad, g1_b, as, ad, menc, zbuf, HEADS, HID, stream);

    // --- GAT layer 2: [30000,1024] -> h[30000,128], 1 head -> bufP[30000,128] ---
    run_gemm(bufP, g2_W, nullptr, bufQ, NNODES, 1024, 128, false, false, stream);
    run_gat_layer(ei, bufQ, bufP, g2_as, g2_ad, g2_b, as, ad, menc, zbuf, 1, HID, stream);

    // --- pool + cross-attention + fusion ---
    pool_kernel<<<3, 128, 0, stream>>>(bufP, pooled, N_PER);
    head_kernel<<<1, 128, 0, stream>>>(pooled, attn_Win, attn_bin, attn_Wout, attn_bout,
                                       fus_W1, fus_b1, fus_W2, fus_b2, outF);
}